// HeteroGraph_59794534695044
// MI455X (gfx1250) — compile-verified
//
#include <hip/hip_runtime.h>

#define HD 128
#define LDSW 136  // 128 + 8 dword pad per row (TDM pad) -> conflict-free DS reads

typedef __attribute__((ext_vector_type(2))) float v2f;
typedef __attribute__((ext_vector_type(8))) float v8f;
typedef __attribute__((ext_vector_type(4))) unsigned v4u;
typedef __attribute__((ext_vector_type(8))) int v8i;
typedef __attribute__((ext_vector_type(4))) int v4i;

// ---------- helpers ----------
__device__ __forceinline__ unsigned f2ord(float f) {
  unsigned u = __float_as_uint(f);
  return (u & 0x80000000u) ? ~u : (u | 0x80000000u);
}
__device__ __forceinline__ float ord2f(unsigned u) {
  return (u & 0x80000000u) ? __uint_as_float(u & 0x7fffffffu) : __uint_as_float(~u);
}
__device__ __forceinline__ float lrelu(float v) { return v > 0.f ? v : 0.2f * v; }

// ---------- GEMM: Hout[n x 128] = X[n x 128] @ W[128 x 128], f32 WMMA ----------
// Block = 256 threads = 8 waves; block covers 64 rows x 128 cols.
// W staged to LDS once per block by the Tensor Data Mover (TDM), with row
// padding (stride 136 dwords) so the two half-wave B reads hit disjoint banks.
// Wave w owns column tile n0 = w*16; B fragments are hoisted to registers and
// reused across 4 row tiles.
__global__ __launch_bounds__(256) void gemm_xW(const float* __restrict__ X,
                                               const float* __restrict__ W,
                                               float* __restrict__ Hout, int nrows) {
  __shared__ float lds_w[HD * LDSW];  // 69,632 B

  // ---- stage W -> LDS via TDM (wave 0 issues; TDM ignores EXEC) ----
  if (threadIdx.x < 32) {
    unsigned lds_base = (unsigned)(uintptr_t)(&lds_w[0]);  // flat addr low 32 = LDS offset
    unsigned long long ga = (unsigned long long)(uintptr_t)W;
    // D# group0: count=1 | lds_addr | global_addr[56:0] | type=2
    v4u g0;
    g0[0] = 1u;                                   // count = 1 valid descriptor
    g0[1] = lds_base;                              // lds_addr (bytes)
    g0[2] = (unsigned)ga;                          // global_addr[31:0]
    g0[3] = (unsigned)(ga >> 32) | (2u << 30);     // global_addr[56:32] | type=2
    // D# group1: data_size=4B, pad every 128 dwords by 8 dwords,
    //            tensor 128x128 (stride0=128), tile 128x128
    v8i g1;
    g1[0] = (int)((2u << 16)        // data_size = 4 bytes
                  | (1u << 20)      // pad_enable
                  | (6u << 22)      // pad_interval: 2^6 * 8B = 128 dwords
                  | (7u << 25));    // pad_amount: 8 dwords
    g1[1] = (int)(128u << 16);      // tensor_dim0[15:0]  (bits 63:48)
    g1[2] = (int)(128u << 16);      // tensor_dim1[15:0]  (bits 95:80); dim0 hi16 = 0
    g1[3] = (int)(128u << 16);      // tile_dim0          (bits 127:112); dim1 hi16 = 0
    g1[4] = 128;                    // tile_dim1          (bits 143:128); tile_dim2 = 0
    g1[5] = 128;                    // tensor_dim0_stride[31:0]
    g1[6] = 0;                      // stride0 hi16, tensor_dim1_stride lo16
    g1[7] = 0;
    v4i gz = {0, 0, 0, 0};          // groups 2/3 unused (2-D tensor)
    v8i gz8 = {0, 0, 0, 0, 0, 0, 0, 0};
    __builtin_amdgcn_tensor_load_to_lds(g0, g1, gz, gz, gz8, 0);
    __builtin_amdgcn_s_wait_tensorcnt(0);
  }
  __syncthreads();
  asm volatile("" ::: "memory");

  const int lane = threadIdx.x & 31;
  const int wave = threadIdx.x >> 5;
  const int n0 = wave * 16;
  const int half = lane >> 4;  // 0: lanes 0-15, 1: lanes 16-31
  const int l15 = lane & 15;

  // Preload B fragments for this wave's column tile from LDS.
  // B 4x16 f32: VGPR j -> K = 2*(lane>>4)+j, N = lane&15
  v2f bfr[32];
#pragma unroll
  for (int kk = 0; kk < 32; ++kk) {
    int k = kk * 4 + 2 * half;
    bfr[kk][0] = lds_w[k * LDSW + n0 + l15];
    bfr[kk][1] = lds_w[(k + 1) * LDSW + n0 + l15];
  }

  for (int rt = 0; rt < 4; ++rt) {
    int m0 = blockIdx.x * 64 + rt * 16;
    if (m0 >= nrows) break;
    v8f c = {0.f, 0.f, 0.f, 0.f, 0.f, 0.f, 0.f, 0.f};
    const float* xrow = X + (size_t)(m0 + l15) * HD;
#pragma unroll
    for (int kk = 0; kk < 32; ++kk) {
      // A 16x4 f32: lane L -> M = L&15 ; VGPR0: K=2*(L>>4), VGPR1: K=2*(L>>4)+1
      v2f a;
      a[0] = xrow[kk * 4 + 2 * half];
      a[1] = xrow[kk * 4 + 2 * half + 1];
      c = __builtin_amdgcn_wmma_f32_16x16x4_f32(false, a, false, bfr[kk], (short)0,
                                                c, false, false);
    }
    // D 16x16 f32: VGPR r -> M = r + 8*(L>>4), N = L&15
#pragma unroll
    for (int r = 0; r < 8; ++r) {
      int m = m0 + r + 8 * half;
      Hout[(size_t)m * HD + n0 + l15] = c[r];
    }
  }
}

// ---------- rowdot: out[i] = dot(Rows[i, 0:128], vec) ; one wave per row ----------
__global__ __launch_bounds__(256) void rowdot128(const float* __restrict__ Rows,
                                                 const float* __restrict__ vec,
                                                 float* __restrict__ out, int n) {
  int node = blockIdx.x * 8 + (threadIdx.x >> 5);
  int lane = threadIdx.x & 31;
  if (node >= n) return;
  const float* row = Rows + (size_t)node * HD;
  float s = 0.f;
#pragma unroll
  for (int i = 0; i < 4; ++i) s += row[lane + 32 * i] * vec[lane + 32 * i];
  for (int off = 16; off; off >>= 1) s += __shfl_down(s, off, 32);
  if (lane == 0) out[node] = s;
}

// ---------- watt[k] = sum_n W[k,n] * att[n] (W @ att) ----------
__global__ void matvec_W_att(const float* __restrict__ W,
                             const float* __restrict__ att,
                             float* __restrict__ out) {
  int k = threadIdx.x;  // 128 threads
  float s = 0.f;
  for (int n = 0; n < HD; ++n) s += W[(size_t)k * HD + n] * att[n];
  out[k] = s;
}

// ---------- fills ----------
__global__ void fill_u32(unsigned* __restrict__ p, unsigned v, int n) {
  int i = blockIdx.x * 256 + threadIdx.x;
  if (i < n) p[i] = v;
}
__global__ void fill_f32(float* __restrict__ p, float v, long long n) {
  long long i = (long long)blockIdx.x * 256 + threadIdx.x;
  if (i < n) p[i] = v;
}

// ---------- input linear: out[n x 128] = X[n x K] @ W[K x 128] + b ----------
__global__ void lin_in(const float* __restrict__ X, const float* __restrict__ W,
                       const float* __restrict__ b, float* __restrict__ out,
                       int n, int K) {
  long long idx = (long long)blockIdx.x * 256 + threadIdx.x;
  if (idx >= (long long)n * HD) return;
  int node = (int)(idx >> 7), ch = (int)(idx & 127);
  float s = b[ch];
  for (int k = 0; k < K; ++k) s += X[(size_t)node * K + k] * W[(size_t)k * HD + ch];
  out[idx] = s;
}

// ---------- edge pass 1: segment max of leaky_relu(a_src[s]+a_dst[d]) ----------
__global__ void edge_max(const int* __restrict__ esrc, const int* __restrict__ edst,
                         const float* __restrict__ a_src,
                         const float* __restrict__ a_dst,
                         unsigned* __restrict__ emax_ord, int ne) {
  int e = blockIdx.x * 256 + threadIdx.x;
  if (e >= ne) return;
  int d = edst[e];
  float v = lrelu(a_src[esrc[e]] + a_dst[d]);
  atomicMax(emax_ord + d, f2ord(v));
}

// ---------- convert ordered-uint max -> float, non-finite -> 0 (in place) ----------
__global__ void finalize_max(unsigned* __restrict__ p, int n) {
  int i = blockIdx.x * 256 + threadIdx.x;
  if (i >= n) return;
  float v = ord2f(p[i]);
  if (!isfinite(v)) v = 0.f;
  ((float*)p)[i] = v;
}

// ---------- edge pass 2: ex = exp(e - emax[d]); denom[d] += ex ----------
__global__ void edge_exp(const int* __restrict__ esrc, const int* __restrict__ edst,
                         const float* __restrict__ a_src,
                         const float* __restrict__ a_dst,
                         const float* __restrict__ emax, float* __restrict__ ex,
                         float* __restrict__ denom, int ne) {
  int e = blockIdx.x * 256 + threadIdx.x;
  if (e >= ne) return;
  int d = edst[e];
  float v = lrelu(a_src[esrc[e]] + a_dst[d]);
  float xv = expf(v - emax[d]);
  ex[e] = xv;
  atomicAdd(denom + d, xv);
}

// ---------- edge pass 3: acc[d] += alpha * h_src[s] ; one wave per edge ----------
__global__ __launch_bounds__(256) void edge_scatter(
    const int* __restrict__ esrc, const int* __restrict__ edst,
    const float* __restrict__ Hsrc, const float* __restrict__ ex,
    const float* __restrict__ denom, float* __restrict__ acc, int ne) {
  int e = blockIdx.x * 8 + (threadIdx.x >> 5);
  int lane = threadIdx.x & 31;
  if (e >= ne) return;
  int s = esrc[e], d = edst[e];
  float alpha = ex[e] / (denom[d] + 1e-16f);
  const float4* hs = (const float4*)(Hsrc + (size_t)s * HD);
  float4 v = hs[lane];
  float* dp = acc + (size_t)d * HD + lane * 4;
  atomicAdd(dp + 0, alpha * v.x);
  atomicAdd(dp + 1, alpha * v.y);
  atomicAdd(dp + 2, alpha * v.z);
  atomicAdd(dp + 3, alpha * v.w);
}

// ---------- per-rel bias add over all dst nodes ----------
__global__ void add_bias(float* __restrict__ acc, const float* __restrict__ bias,
                         long long n128) {
  long long i = (long long)blockIdx.x * 256 + threadIdx.x;
  if (i < n128) acc[i] += bias[i & 127];
}

// ---------- x = relu(v)+v ----------
__global__ void act_relu_plus(float* __restrict__ x, const float* __restrict__ acc,
                              long long n) {
  long long i = (long long)blockIdx.x * 256 + threadIdx.x;
  if (i >= n) return;
  float v = acc[i];
  x[i] = v > 0.f ? v + v : v;
}

// ---------- mean pool over operator nodes ----------
__global__ __launch_bounds__(256) void pool_sum(const float* __restrict__ xop,
                                                const int* __restrict__ batch,
                                                float* __restrict__ psum,
                                                float* __restrict__ pcnt, int n) {
  int node = blockIdx.x * 8 + (threadIdx.x >> 5);
  int lane = threadIdx.x & 31;
  if (node >= n) return;
  int g = batch[node];
  const float4* r = (const float4*)(xop + (size_t)node * HD);
  float4 v = r[lane];
  float* dp = psum + (size_t)g * HD + lane * 4;
  atomicAdd(dp + 0, v.x);
  atomicAdd(dp + 1, v.y);
  atomicAdd(dp + 2, v.z);
  atomicAdd(dp + 3, v.w);
  if (lane == 0) atomicAdd(pcnt + g, 1.0f);
}

// ---------- out[g] = (psum[g]/max(cnt,1)) . W + b ; one wave per graph ----------
__global__ __launch_bounds__(256) void final_lin(const float* __restrict__ psum,
                                                 const float* __restrict__ pcnt,
                                                 const float* __restrict__ W,
                                                 const float* __restrict__ b,
                                                 float* __restrict__ out, int ng) {
  int g = blockIdx.x * 8 + (threadIdx.x >> 5);
  int lane = threadIdx.x & 31;
  if (g >= ng) return;
  float c = pcnt[g];
  if (!(c > 1.f)) c = 1.f;
  const float* row = psum + (size_t)g * HD;
  float s = 0.f;
#pragma unroll
  for (int i = 0; i < 4; ++i) s += row[lane + 32 * i] * W[lane + 32 * i];
  for (int off = 16; off; off >>= 1) s += __shfl_down(s, off, 32);
  if (lane == 0) out[g] = s / c + b[0];
}

// ============================ host orchestration ============================
extern "C" void kernel_launch(void* const* d_in, const int* in_sizes, int n_in,
                              void* d_out, int out_size, void* d_ws, size_t ws_size,
                              hipStream_t stream) {
  // node types: 0=operator 1=table 2=column 3=predicate
  static const int NT[4] = {200000, 50000, 300000, 100000};
  static const int IND[4] = {4, 2, 16, 4};
  struct Rel { int eIdx, E, srcT, dstT, sortIdx; };
  // params pytree flatten order (sorted dict keys):
  // conv1[calledby,connects,filters,outputby,scannedby,selfloop_column,selfloop_table]
  //   each: W, att_dst, att_src, bias   (13..40), conv2 (41..68),
  // lin: W=69 b=70, lin_column: 71/72, lin_operator: 73/74,
  // lin_predicate: 75/76, lin_table: 77/78
  static const Rel rels[7] = {
      {4, 200000, 1, 0, 4},   // scannedby: table->operator
      {5, 300000, 3, 0, 2},   // filters: predicate->operator
      {6, 600000, 2, 0, 3},   // outputby: column->operator
      {7, 300000, 2, 3, 1},   // connects: column->predicate
      {8, 400000, 0, 0, 0},   // calledby: operator->operator
      {9, 50000, 1, 1, 6},    // selfloop_table
      {10, 300000, 2, 2, 5},  // selfloop_column
  };

  float* ws = (float*)d_ws;
  size_t off = 0;
  auto alloc = [&](size_t nf) { float* p = ws + off; off += nf; return p; };

  float* x[4];
  float* acc[4];
  for (int t = 0; t < 4; ++t) x[t] = alloc((size_t)NT[t] * HD);
  for (int t = 0; t < 4; ++t) acc[t] = alloc((size_t)NT[t] * HD);
  float* h_src = alloc((size_t)300000 * HD);
  float* a_src = alloc(300000);
  float* a_dst = alloc(300000);
  float* emaxb = alloc(300000);  // unsigned during max pass, float after
  float* denom = alloc(300000);
  float* exbuf = alloc(600000);
  float* watt = alloc(HD);
  float* psum = alloc((size_t)4096 * HD);
  float* pcnt = alloc(4096);
  (void)ws_size; (void)in_sizes; (void)n_in;

  // --- input linear layers ---
  static const int linWi[4] = {73, 77, 71, 75};
  static const int linBi[4] = {74, 78, 72, 76};
  for (int t = 0; t < 4; ++t) {
    long long n128 = (long long)NT[t] * HD;
    lin_in<<<dim3((unsigned)((n128 + 255) / 256)), 256, 0, stream>>>(
        (const float*)d_in[t], (const float*)d_in[linWi[t]],
        (const float*)d_in[linBi[t]], x[t], NT[t], IND[t]);
  }

  // --- two hetero GAT conv layers ---
  for (int layer = 0; layer < 2; ++layer) {
    int base = 13 + layer * 28;
    for (int t = 0; t < 4; ++t) {
      long long n128 = (long long)NT[t] * HD;
      fill_f32<<<dim3((unsigned)((n128 + 255) / 256)), 256, 0, stream>>>(acc[t], 0.f, n128);
    }
    for (int r = 0; r < 7; ++r) {
      const Rel& R = rels[r];
      const float* Wp = (const float*)d_in[base + R.sortIdx * 4 + 0];
      const float* attD = (const float*)d_in[base + R.sortIdx * 4 + 1];
      const float* attS = (const float*)d_in[base + R.sortIdx * 4 + 2];
      const float* bias = (const float*)d_in[base + R.sortIdx * 4 + 3];
      const int* esrc = (const int*)d_in[R.eIdx];
      const int* edst = esrc + R.E;
      int ns = NT[R.srcT], nd = NT[R.dstT];

      // h_src = x_src @ W   (f32 WMMA, W staged via TDM)
      gemm_xW<<<dim3((ns + 63) / 64), 256, 0, stream>>>(x[R.srcT], Wp, h_src, ns);
      // a_src = h_src @ att_src
      rowdot128<<<dim3((ns + 7) / 8), 256, 0, stream>>>(h_src, attS, a_src, ns);
      // a_dst = x_dst @ (W @ att_dst)   -- skip materializing h_dst
      matvec_W_att<<<dim3(1), 128, 0, stream>>>(Wp, attD, watt);
      rowdot128<<<dim3((nd + 7) / 8), 256, 0, stream>>>(x[R.dstT], watt, a_dst, nd);

      // segment softmax over dst
      fill_u32<<<dim3((nd + 255) / 256), 256, 0, stream>>>((unsigned*)emaxb,
                                                           0x007FFFFFu, nd);  // ord(-inf)
      fill_u32<<<dim3((nd + 255) / 256), 256, 0, stream>>>((unsigned*)denom, 0u, nd);
      edge_max<<<dim3((R.E + 255) / 256), 256, 0, stream>>>(esrc, edst, a_src, a_dst,
                                                            (unsigned*)emaxb, R.E);
      finalize_max<<<dim3((nd + 255) / 256), 256, 0, stream>>>((unsigned*)emaxb, nd);
      edge_exp<<<dim3((R.E + 255) / 256), 256, 0, stream>>>(esrc, edst, a_src, a_dst,
                                                            emaxb, exbuf, denom, R.E);
      // acc[dst] += alpha * h_src[src]
      edge_scatter<<<dim3((R.E + 7) / 8), 256, 0, stream>>>(esrc, edst, h_src, exbuf,
                                                            denom, acc[R.dstT], R.E);
      long long nd128 = (long long)nd * HD;
      add_bias<<<dim3((unsigned)((nd128 + 255) / 256)), 256, 0, stream>>>(acc[R.dstT],
                                                                          bias, nd128);
    }
    for (int t = 0; t < 4; ++t) {
      long long n128 = (long long)NT[t] * HD;
      act_relu_plus<<<dim3((unsigned)((n128 + 255) / 256)), 256, 0, stream>>>(
          x[t], acc[t], n128);
    }
  }

  // --- global mean pool over operator nodes + final linear ---
  int ng = out_size;  // 4096
  long long np = (long long)ng * HD;
  fill_f32<<<dim3((unsigned)((np + 255) / 256)), 256, 0, stream>>>(psum, 0.f, np);
  fill_f32<<<dim3((ng + 255) / 256), 256, 0, stream>>>(pcnt, 0.f, ng);
  pool_sum<<<dim3((NT[0] + 7) / 8), 256, 0, stream>>>(x[0], (const int*)d_in[11], psum,
                                                      pcnt, NT[0]);
  final_lin<<<dim3((ng + 7) / 8), 256, 0, stream>>>(psum, pcnt, (const float*)d_in[69],
                                                    (const float*)d_in[70],
                                                    (float*)d_out, ng);
}